// GraphSoftmax_42305427866218
// MI455X (gfx1250) — compile-verified
//
#include <hip/hip_runtime.h>
#include <math.h>

// ---------------------------------------------------------------------------
// GraphSoftmax (segmented softcap-softmax, sorted index), MI455X / gfx1250.
//
// Memory-bound: ~630 MB min HBM traffic -> ~27 us @ 23.3 TB/s. Design goal:
// touch src / exp_rescale / out exactly once in HBM, do all re-reads from LDS
// (staged with CDNA5 async global->LDS), and avoid atomics entirely by
// exploiting the sorted index (wave-per-node segmented reduction in regs).
// ---------------------------------------------------------------------------

#define SOFT_CAP   30.0f
#define EPS_F      1e-16f
#define HFEAT      16
#define NNODES     100000
#define CAPE       128        // max edges staged in LDS per node (8 KB / wave)
#define WPB        8          // waves (= nodes) per 256-thread block

typedef __attribute__((ext_vector_type(4))) int v4i;

// ---- CDNA5 async global->LDS (tracked by ASYNCcnt) ------------------------
// Toolchain probe: builtin takes int4* pointers (generic), per hipcc diag.
__device__ __forceinline__ void stage_b128(const float4* g, float4* l) {
#if __has_builtin(__builtin_amdgcn_global_load_async_to_lds_b128)
  __builtin_amdgcn_global_load_async_to_lds_b128((v4i*)g, (v4i*)l, 0, 0);
#else
  *l = *g;  // synchronous fallback
#endif
}

__device__ __forceinline__ void wait_async0() {
#if __has_builtin(__builtin_amdgcn_s_wait_asynccnt)
  __builtin_amdgcn_s_wait_asynccnt(0);
#else
  asm volatile("s_wait_asynccnt 0" ::: "memory");
#endif
}

// ---- math helpers ---------------------------------------------------------
__device__ __forceinline__ float softcap(float x) {
  return tanhf(x * (1.0f / SOFT_CAP)) * SOFT_CAP;
}

// Reduce across the 8 lanes {g, g+4, ..., g+28} that share feature group
// g = lane & 3 (wave32 xor tree). Result replicated on all lanes of a group.
__device__ __forceinline__ float4 group_reduce_max(float4 v) {
#pragma unroll
  for (int m = 4; m <= 16; m <<= 1) {
    v.x = fmaxf(v.x, __shfl_xor(v.x, m, 32));
    v.y = fmaxf(v.y, __shfl_xor(v.y, m, 32));
    v.z = fmaxf(v.z, __shfl_xor(v.z, m, 32));
    v.w = fmaxf(v.w, __shfl_xor(v.w, m, 32));
  }
  return v;
}

__device__ __forceinline__ float4 group_reduce_sum(float4 v) {
#pragma unroll
  for (int m = 4; m <= 16; m <<= 1) {
    v.x += __shfl_xor(v.x, m, 32);
    v.y += __shfl_xor(v.y, m, 32);
    v.z += __shfl_xor(v.z, m, 32);
    v.w += __shfl_xor(v.w, m, 32);
  }
  return v;
}

// ---------------------------------------------------------------------------
// Kernel 1: sorted index -> CSR offsets. off[n] = first edge with index >= n,
// off[N] = E. Each boundary written exactly once.
// ---------------------------------------------------------------------------
__global__ __launch_bounds__(256) void gs_build_offsets(
    const int* __restrict__ index, int* __restrict__ off, int E, int N) {
  int e = blockIdx.x * blockDim.x + threadIdx.x;
  if (e >= E) return;
  int cur  = index[e];
  int prev = (e == 0) ? -1 : index[e - 1];
  for (int n = prev + 1; n <= cur; ++n) off[n] = e;
  if (e == E - 1) {
    for (int n = cur + 1; n <= N; ++n) off[n] = E;  // trailing empty nodes
  }
}

// ---------------------------------------------------------------------------
// Kernel 2: one wave32 per node. Lane l owns float4 chunks i = l, l+32, ...
// of the segment; chunk i covers edge (start + i/4), features 4*(i%4)..+3.
// Since 32 % 4 == 0, a lane's feature group is fixed at (l & 3).
// ---------------------------------------------------------------------------
__global__ __launch_bounds__(WPB * 32) void gs_softmax(
    const float4* __restrict__ src, const float4* __restrict__ rsc,
    const int* __restrict__ off, float4* __restrict__ out, int nnodes) {
  __shared__ float4 stage[WPB * CAPE * 4];  // 64 KB: 8 KB wave-private slabs

  const int lane = threadIdx.x & 31;
  const int wid  = threadIdx.x >> 5;
  const int node = blockIdx.x * WPB + wid;
  if (node >= nnodes) return;

  const int s0  = off[node];
  const int s1  = off[node + 1];
  const int len = s1 - s0;
  if (len <= 0) return;              // empty node: no edges, nothing to write

  const int nch  = len * 4;          // float4 chunks in this segment
  const int base = s0 * 4;           // float4 index of segment start

  float4 mx = make_float4(-__builtin_inff(), -__builtin_inff(),
                          -__builtin_inff(), -__builtin_inff());
  float4 sm = make_float4(0.f, 0.f, 0.f, 0.f);

  if (len <= CAPE) {
    // ---------- fast path: segment staged once in LDS ----------
    float4* w = &stage[wid * (CAPE * 4)];

    // Async DMA the raw src segment into this wave's LDS slab.
    for (int i = lane; i < nch; i += 32) stage_b128(&src[base + i], &w[i]);
    wait_async0();

    // Pass 1: softcap in place, track per-feature max.
    for (int i = lane; i < nch; i += 32) {
      float4 v = w[i];
      v.x = softcap(v.x); v.y = softcap(v.y);
      v.z = softcap(v.z); v.w = softcap(v.w);
      w[i] = v;
      mx.x = fmaxf(mx.x, v.x); mx.y = fmaxf(mx.y, v.y);
      mx.z = fmaxf(mx.z, v.z); mx.w = fmaxf(mx.w, v.w);
    }
    mx = group_reduce_max(mx);

    // Pass 2: exp(s - max) * rescale (rescale streamed from HBM once),
    // overwrite LDS with unnormalized out, accumulate per-feature sum.
    for (int i = lane; i < nch; i += 32) {
      float4 s = w[i];
      float4 r = rsc[base + i];
      float4 o;
      o.x = expf(s.x - mx.x) * r.x;
      o.y = expf(s.y - mx.y) * r.y;
      o.z = expf(s.z - mx.z) * r.z;
      o.w = expf(s.w - mx.w) * r.w;
      w[i] = o;
      sm.x += o.x; sm.y += o.y; sm.z += o.z; sm.w += o.w;
    }
    sm = group_reduce_sum(sm);
    float4 inv = make_float4(1.f / (sm.x + EPS_F), 1.f / (sm.y + EPS_F),
                             1.f / (sm.z + EPS_F), 1.f / (sm.w + EPS_F));

    // Pass 3: normalize and write out (single coalesced b128 store stream).
    for (int i = lane; i < nch; i += 32) {
      float4 o = w[i];
      o.x *= inv.x; o.y *= inv.y; o.z *= inv.z; o.w *= inv.w;
      out[base + i] = o;
    }
  } else {
    // ---------- rare fallback: segment larger than LDS slab ----------
    for (int i = lane; i < nch; i += 32) {
      float4 v = src[base + i];
      v.x = softcap(v.x); v.y = softcap(v.y);
      v.z = softcap(v.z); v.w = softcap(v.w);
      mx.x = fmaxf(mx.x, v.x); mx.y = fmaxf(mx.y, v.y);
      mx.z = fmaxf(mx.z, v.z); mx.w = fmaxf(mx.w, v.w);
    }
    mx = group_reduce_max(mx);

    for (int i = lane; i < nch; i += 32) {
      float4 v = src[base + i];
      float4 r = rsc[base + i];
      float4 o;
      o.x = expf(softcap(v.x) - mx.x) * r.x;
      o.y = expf(softcap(v.y) - mx.y) * r.y;
      o.z = expf(softcap(v.z) - mx.z) * r.z;
      o.w = expf(softcap(v.w) - mx.w) * r.w;
      out[base + i] = o;   // use our own output buffer as scratch
      sm.x += o.x; sm.y += o.y; sm.z += o.z; sm.w += o.w;
    }
    sm = group_reduce_sum(sm);
    float4 inv = make_float4(1.f / (sm.x + EPS_F), 1.f / (sm.y + EPS_F),
                             1.f / (sm.z + EPS_F), 1.f / (sm.w + EPS_F));

    for (int i = lane; i < nch; i += 32) {
      float4 o = out[base + i];
      o.x *= inv.x; o.y *= inv.y; o.z *= inv.z; o.w *= inv.w;
      out[base + i] = o;
    }
  }
}

// ---------------------------------------------------------------------------
extern "C" void kernel_launch(void* const* d_in, const int* in_sizes, int n_in,
                              void* d_out, int out_size, void* d_ws, size_t ws_size,
                              hipStream_t stream) {
  const float4* src = (const float4*)d_in[0];   // [E,16] f32
  const float4* rsc = (const float4*)d_in[1];   // [E,16] f32
  const int*    idx = (const int*)d_in[2];      // [E] i32, sorted
  (void)n_in; (void)ws_size; (void)out_size;

  const int E = in_sizes[0] / HFEAT;            // 3,200,000
  const int N = NNODES;                         // reference constant

  int* off = (int*)d_ws;                        // (N+1) ints of scratch

  gs_build_offsets<<<(E + 255) / 256, 256, 0, stream>>>(idx, off, E, N);
  gs_softmax<<<(N + WPB - 1) / WPB, WPB * 32, 0, stream>>>(
      src, rsc, off, (float4*)d_out, N);
}